// CapsuleLayer_62457414418698
// MI455X (gfx1250) — compile-verified
//
#include <hip/hip_runtime.h>

typedef float v2f __attribute__((ext_vector_type(2)));
typedef float v8f __attribute__((ext_vector_type(8)));

#define BATCH    128
#define IN_CAPS  1152
#define IN_DIM   8
#define N_CAPS   16
#define DIM      16
#define ROUTINGS 3

// ---------------------------------------------------------------------------
// Phase 1: hat[b][n][i][e] = sum_d x[b][i][d] * W[n][i][e][d]
// Per (16-batch tile, n, i): two chained V_WMMA_F32_16X16X4_F32 (K=0..3, 4..7).
// A = x tile (M=batch 16, K=d), B = W tile (K=d, N=e 16), C/D f32 16x16.
// ---------------------------------------------------------------------------
__global__ __launch_bounds__(256)
void caps_hat_wmma(const float* __restrict__ x,
                   const float* __restrict__ W,
                   float* __restrict__ hat) {
    const int lane = threadIdx.x & 31;
    const int wave = threadIdx.x >> 5;       // 0..7
    const int col  = lane & 15;              // N / M-low index
    const int h    = lane >> 4;              // half-wave select
    const int b0   = blockIdx.x * 16;        // batch tile base (8 tiles)
    const int n    = blockIdx.y;             // output capsule (16)
    const int iz   = blockIdx.z;             // i-slice (6)

    // Per-lane source rows (ISA 7.12.2 layouts for 32-bit A/B):
    //   A reg r : x[b0+col][i][2h + r]      (K = 2h + r, +k0)
    //   B reg r : W[n][i][col][2h + r]      (K = 2h + r, +k0)
    const float* xrow  = x + (size_t)(b0 + col) * IN_CAPS * IN_DIM;
    const float* wbase = W + (size_t)n * IN_CAPS * (DIM * IN_DIM);

    for (int i = iz * 8 + wave; i < IN_CAPS; i += 48) {
        const float* xp = xrow  + (size_t)i * IN_DIM;
        const float* wp = wbase + ((size_t)i * DIM + col) * IN_DIM;

        v2f a0 = *(const v2f*)(xp + 2 * h);          // K = 2h, 2h+1
        v2f a1 = *(const v2f*)(xp + 4 + 2 * h);      // K = 4+2h, 5+2h
        v2f bb0 = *(const v2f*)(wp + 2 * h);
        v2f bb1 = *(const v2f*)(wp + 4 + 2 * h);

        v8f c = {};
        c = __builtin_amdgcn_wmma_f32_16x16x4_f32(false, a0, false, bb0,
                                                  (short)0, c, false, false);
        c = __builtin_amdgcn_wmma_f32_16x16x4_f32(false, a1, false, bb1,
                                                  (short)0, c, false, false);

        // D layout: lane(h,col) reg r -> M = r + 8h, N = col
        #pragma unroll
        for (int r = 0; r < 8; ++r) {
            const int brow = b0 + r + 8 * h;
            hat[(((size_t)brow * N_CAPS + n) * IN_CAPS + i) * DIM + col] = c[r];
        }
    }
}

// ---------------------------------------------------------------------------
// Phase 2: dynamic routing, one workgroup per batch element.
// logits + softmax probs live in LDS (2 x 73.7 KB < 320 KB WGP LDS).
// hat (151 MB total) is L2-resident; each block re-reads its 1.18 MB slice.
// ---------------------------------------------------------------------------
__global__ __launch_bounds__(256)
void caps_route(const float* __restrict__ hat,
                const float* __restrict__ Bb,
                float* __restrict__ out) {
    __shared__ float logits[N_CAPS][IN_CAPS];   // 73728 B
    __shared__ float cprob [N_CAPS][IN_CAPS];   // 73728 B
    __shared__ float sbuf  [N_CAPS][DIM];
    __shared__ float obuf  [N_CAPS][DIM];

    const int t = threadIdx.x;                  // 0..255
    const int b = blockIdx.x;                   // batch element
    const float* hb = hat + (size_t)b * N_CAPS * IN_CAPS * DIM;

    for (int idx = t; idx < N_CAPS * IN_CAPS; idx += 256)
        (&logits[0][0])[idx] = 0.0f;
    __syncthreads();

    const int n = t >> 4;                       // 0..15
    const int e = t & 15;                       // 0..15

    for (int it = 0; it < ROUTINGS; ++it) {
        // --- softmax over n for each input capsule i ---
        for (int i = t; i < IN_CAPS; i += 256) {
            float m = logits[0][i];
            #pragma unroll
            for (int k = 1; k < N_CAPS; ++k) m = fmaxf(m, logits[k][i]);
            float ex[N_CAPS];
            float z = 0.0f;
            #pragma unroll
            for (int k = 0; k < N_CAPS; ++k) {
                ex[k] = __expf(logits[k][i] - m);
                z += ex[k];
            }
            const float inv = 1.0f / z;
            #pragma unroll
            for (int k = 0; k < N_CAPS; ++k) cprob[k][i] = ex[k] * inv;
        }
        __syncthreads();

        // --- s[n][e] = sum_i c[n][i] * hat[n][i][e] + Bb[n][e] ---
        {
            float acc = 0.0f;
            const float* hne = hb + (size_t)n * IN_CAPS * DIM + e;
            for (int i = 0; i < IN_CAPS; ++i)
                acc = fmaf(cprob[n][i], hne[(size_t)i * DIM], acc);
            sbuf[n][e] = acc + Bb[n * DIM + e];
        }
        __syncthreads();

        // --- squash per capsule n ---
        {
            float ss = 0.0f;
            #pragma unroll
            for (int k = 0; k < DIM; ++k) {
                const float v = sbuf[n][k];
                ss = fmaf(v, v, ss);
            }
            const float scale = ss / (1.0f + ss) * rsqrtf(ss);
            obuf[n][e] = scale * sbuf[n][e];
        }
        __syncthreads();

        if (it == ROUTINGS - 1) break;

        // --- logits[n][i] += sum_e out[n][e] * hat[n][i][e] ---
        for (int p = t; p < N_CAPS * IN_CAPS; p += 256) {
            const int nn = p / IN_CAPS;
            const int ii = p % IN_CAPS;
            const float4* hp4 =
                (const float4*)(hb + ((size_t)nn * IN_CAPS + ii) * DIM);
            const float4 q0 = hp4[0], q1 = hp4[1], q2 = hp4[2], q3 = hp4[3];
            float d = 0.0f;
            d = fmaf(obuf[nn][0],  q0.x, d); d = fmaf(obuf[nn][1],  q0.y, d);
            d = fmaf(obuf[nn][2],  q0.z, d); d = fmaf(obuf[nn][3],  q0.w, d);
            d = fmaf(obuf[nn][4],  q1.x, d); d = fmaf(obuf[nn][5],  q1.y, d);
            d = fmaf(obuf[nn][6],  q1.z, d); d = fmaf(obuf[nn][7],  q1.w, d);
            d = fmaf(obuf[nn][8],  q2.x, d); d = fmaf(obuf[nn][9],  q2.y, d);
            d = fmaf(obuf[nn][10], q2.z, d); d = fmaf(obuf[nn][11], q2.w, d);
            d = fmaf(obuf[nn][12], q3.x, d); d = fmaf(obuf[nn][13], q3.y, d);
            d = fmaf(obuf[nn][14], q3.z, d); d = fmaf(obuf[nn][15], q3.w, d);
            logits[nn][ii] += d;
        }
        __syncthreads();
    }

    // out[b][n][e]
    out[(size_t)b * N_CAPS * DIM + t] = obuf[n][e];
}

// ---------------------------------------------------------------------------
extern "C" void kernel_launch(void* const* d_in, const int* in_sizes, int n_in,
                              void* d_out, int out_size, void* d_ws, size_t ws_size,
                              hipStream_t stream) {
    (void)in_sizes; (void)n_in; (void)out_size; (void)ws_size;
    const float* x  = (const float*)d_in[0];   // [128,1152,8]
    const float* W  = (const float*)d_in[1];   // [16,1152,16,8]
    const float* Bb = (const float*)d_in[2];   // [16,16]
    float* out = (float*)d_out;                // [128,16,16]
    float* hat = (float*)d_ws;                 // [128,16,1152,16] = 151 MB scratch

    dim3 g1(BATCH / 16, N_CAPS, 6);            // 8 x 16 x 6 = 768 workgroups
    caps_hat_wmma<<<g1, 256, 0, stream>>>(x, W, hat);
    caps_route<<<BATCH, 256, 0, stream>>>(hat, Bb, out);
}